// CoreAttention_31327491457419
// MI455X (gfx1250) — compile-verified
//
#include <hip/hip_runtime.h>
#include <math.h>

// Sparse graph attention, B=1, N=50000, H=8, D=16, E=800000, fp32 throughout.
#define HEADS 8
#define DIM 16
#define HD 128          // HEADS*DIM
#define INV_SCALE 0.25f // 1/sqrt(16)

typedef __attribute__((ext_vector_type(2))) float v2f;
typedef __attribute__((ext_vector_type(8))) float v8f;

__global__ void zero_kernel(float* __restrict__ out, float* __restrict__ zbuf,
                            int nOut, int nZ) {
  int i = blockIdx.x * blockDim.x + threadIdx.x;
  if (i < nOut) out[i] = 0.0f;
  if (i < nZ)  zbuf[i] = 0.0f;
}

// One wave32 per group of 16 edges. Scores computed as the diagonal of
// S = K_src * Q_dst^T via 4x v_wmma_f32_16x16x4_f32 per head (full fp32).
__global__ void __launch_bounds__(256)
edge_attn_kernel(const float* __restrict__ q, const float* __restrict__ k,
                 const float* __restrict__ v, const int* __restrict__ eidx,
                 float* __restrict__ wV, float* __restrict__ Z, int nEdges) {
  __shared__ int   s_idx[8][32];  // [wave][0..15]=src, [16..31]=dst
  __shared__ float s_sc[8][16];   // per-wave per-head scores

  const int lane = threadIdx.x & 31;
  const int wv   = threadIdx.x >> 5;
  const long e0  = ((long)blockIdx.x * 8 + wv) * 16;
  if (e0 >= nEdges) return;

  // Stage this group's 16 src + 16 dst indices into LDS.
  {
    long e   = e0 + (lane & 15);
    int  ok  = e < (long)nEdges;
    long off = (lane < 16) ? e : ((long)nEdges + e);
    s_idx[wv][lane] = ok ? eidx[off] : 0;
  }
  asm volatile("s_wait_dscnt 0x0" ::: "memory");

  // Matrix-phase mapping: lane m / m+16 supply edge m's K-chunks (A) and Q-chunks (B).
  const int m     = lane & 15;
  const int is_hi = lane >> 4;
  const int srcm  = s_idx[wv][m];
  const int dstm  = s_idx[wv][16 + m];

  // Scatter-phase mapping: 2 lanes per edge, 8 dims each.
  const int em   = lane >> 1;
  const int half = lane & 1;
  const int srce = s_idx[wv][em];
  const int dste = s_idx[wv][16 + em];
  const bool ev  = (e0 + em) < (long)nEdges;

  const float* arow = k + (long)srcm * HD + is_hi * 2;  // A: K_src fragment base
  const float* brow = q + (long)dstm * HD + is_hi * 2;  // B: Q_dst fragment base
  const float* vrow = v + (long)srce * HD + half * 8;
  float*       orow = wV + (long)dste * HD + half * 8;

  const int  idxsel = lane & 7;
  const bool isdiag = (lane < 8) || (lane >= 24);
  const int  dm     = (lane >= 24) ? (lane - 16) : lane;  // diag slot this lane owns

  for (int h = 0; h < HEADS; ++h) {
    const float* ap = arow + h * DIM;
    const float* bp = brow + h * DIM;
    v8f c = {0.f, 0.f, 0.f, 0.f, 0.f, 0.f, 0.f, 0.f};
#pragma unroll
    for (int c4 = 0; c4 < 4; ++c4) {
      v2f a = *(const v2f*)(ap + c4 * 4);
      v2f b = *(const v2f*)(bp + c4 * 4);
      // D = A(16x4 f32) * B(4x16 f32) + C, accumulate over K chunks.
      c = __builtin_amdgcn_wmma_f32_16x16x4_f32(false, a, false, b,
                                                (short)0, c, false, false);
    }
    // Extract c[idxsel] with a cndmask select tree (diagonal of S).
    float d01 = (idxsel & 1) ? c[1] : c[0];
    float d23 = (idxsel & 1) ? c[3] : c[2];
    float d45 = (idxsel & 1) ? c[5] : c[4];
    float d67 = (idxsel & 1) ? c[7] : c[6];
    float d03 = (idxsel & 2) ? d23 : d01;
    float d47 = (idxsel & 2) ? d67 : d45;
    float dd  = (idxsel & 4) ? d47 : d03;
    if (isdiag) {
      float t = dd * INV_SCALE;
      t = fminf(fmaxf(t, -5.0f), 5.0f);
      s_sc[wv][dm] = __expf(t);
    }
    asm volatile("s_wait_dscnt 0x0" ::: "memory");
    float sc = s_sc[wv][em];

    if (ev) {
      const float4* vp = (const float4*)(vrow + h * DIM);
      float4 p0 = vp[0];
      float4 p1 = vp[1];
      float* op = orow + h * DIM;
      atomicAdd(op + 0, p0.x * sc); atomicAdd(op + 1, p0.y * sc);
      atomicAdd(op + 2, p0.z * sc); atomicAdd(op + 3, p0.w * sc);
      atomicAdd(op + 4, p1.x * sc); atomicAdd(op + 5, p1.y * sc);
      atomicAdd(op + 6, p1.z * sc); atomicAdd(op + 7, p1.w * sc);
      if (half == 0) atomicAdd(Z + (long)dste * HEADS + h, sc);
    }
    // s_sc reads above are in-order with next iteration's LDS store (same wave).
  }
}

__global__ void norm_kernel(float* __restrict__ out, const float* __restrict__ Z,
                            int n) {
  int i = blockIdx.x * blockDim.x + threadIdx.x;
  if (i >= n) return;
  int node = i >> 7;        // / 128
  int h    = (i >> 4) & 7;  // (i % 128) / 16
  out[i] = out[i] / (Z[node * HEADS + h] + 1e-6f);
}

extern "C" void kernel_launch(void* const* d_in, const int* in_sizes, int n_in,
                              void* d_out, int out_size, void* d_ws, size_t ws_size,
                              hipStream_t stream) {
  const float* q    = (const float*)d_in[0];
  const float* k    = (const float*)d_in[1];
  const float* v    = (const float*)d_in[2];
  const int*   eidx = (const int*)d_in[3];
  float* out = (float*)d_out;
  float* Z   = (float*)d_ws;            // N*H fp32 = 1.6 MB of workspace

  const int nEdges = in_sizes[3] / 2;   // edge_index is [2, E]
  const int nOut   = out_size;          // N*H*D
  const int nZ     = nOut / DIM;        // N*H

  const int threads = 256;
  const int blocks0 = (nOut + threads - 1) / threads;
  zero_kernel<<<blocks0, threads, 0, stream>>>(out, Z, nOut, nZ);

  const int groups  = (nEdges + 15) / 16;     // one wave32 per group
  const int blocks1 = (groups + 7) / 8;       // 8 waves per 256-thread block
  edge_attn_kernel<<<blocks1, threads, 0, stream>>>(q, k, v, eidx, out, Z, nEdges);

  norm_kernel<<<blocks0, threads, 0, stream>>>(out, Z, nOut);
}